// AggFeatureSeqEncoder_4956392259659
// MI455X (gfx1250) — compile-verified
//
#include <hip/hip_runtime.h>
#include <hip/hip_bf16.h>

#define T_LEN 2048
#define VM 256
#define VT 128
#define NCOLS 1158
#define EPSF 1e-9f

// CDNA5 async global->LDS load (GVS mode): per-lane,
//   each lane copies 16B from global (base + per-lane offset) to its LDS address.
// IOFFSET applies to BOTH sides (CDNA5 ISA), which we exploit
// to reuse one LDS-address/global-offset VGPR pair for both 4KB-apart chunks.
// Tracked by ASYNCcnt.
__device__ __forceinline__ void async_b128(unsigned lds_off, unsigned goff,
                                           const void* base, int imm) {
  asm volatile("global_load_async_to_lds_b128 %0, %1, %2 offset:%3"
               :: "v"(lds_off), "v"(goff), "s"(base), "i"(imm) : "memory");
}

__device__ __forceinline__ void wait_async0() {
  asm volatile("s_wait_asynccnt 0" ::: "memory");
}

__global__ __launch_bounds__(256) void agg_feature_kernel(
    const float* __restrict__ amount, const int* __restrict__ mcc,
    const int* __restrict__ trt, const int* __restrict__ seq_lens,
    float* __restrict__ out)
{
  __shared__ __align__(16) float s_amt[T_LEN];
  __shared__ __align__(16) int   s_mcc[T_LEN];
  __shared__ __align__(16) int   s_trt[T_LEN];
  __shared__ float h_m_cnt[VM], h_m_sv[VM], h_m_sv2[VM];
  __shared__ float h_t_cnt[VT], h_t_sv[VT], h_t_sv2[VT];
  __shared__ float s_red[8], s_red2[8];   // one partial per wave32
  __shared__ float s_dist[2];

  const int t = threadIdx.x;
  const int b = blockIdx.x;

  const float* arow = amount + (size_t)b * T_LEN;
  const int*   mrow = mcc    + (size_t)b * T_LEN;
  const int*   trow = trt    + (size_t)b * T_LEN;

  // Async-stage the whole 24KB row into LDS: 2 chunks x 3 arrays per thread.
  const unsigned goff  = (unsigned)t * 16u;
  const unsigned l_amt = (unsigned)(size_t)&s_amt[4 * t];
  const unsigned l_mcc = (unsigned)(size_t)&s_mcc[4 * t];
  const unsigned l_trt = (unsigned)(size_t)&s_trt[4 * t];
  async_b128(l_amt, goff, arow, 0);
  async_b128(l_amt, goff, arow, 4096);
  async_b128(l_mcc, goff, mrow, 0);
  async_b128(l_mcc, goff, mrow, 4096);
  async_b128(l_trt, goff, trow, 0);
  async_b128(l_trt, goff, trow, 4096);

  // Overlap histogram init with the async DMA.
  h_m_cnt[t] = 0.f; h_m_sv[t] = 0.f; h_m_sv2[t] = 0.f;
  if (t < VT) { h_t_cnt[t] = 0.f; h_t_sv[t] = 0.f; h_t_sv2[t] = 0.f; }
  if (t < 2) s_dist[t] = 0.f;

  wait_async0();
  __syncthreads();

  // Process 8 elements/thread from LDS; scatter into shared histograms.
  float psum = 0.f, psum2 = 0.f;
#pragma unroll
  for (int k = 0; k < 2; ++k) {
    const int idx4 = t + k * 256;
    const float4 a4 = ((const float4*)s_amt)[idx4];
    const int4   m4 = ((const int4*)s_mcc)[idx4];
    const int4   r4 = ((const int4*)s_trt)[idx4];
    const float av[4] = {a4.x, a4.y, a4.z, a4.w};
    const int   mv[4] = {m4.x, m4.y, m4.z, m4.w};
    const int   rv[4] = {r4.x, r4.y, r4.z, r4.w};
#pragma unroll
    for (int j = 0; j < 4; ++j) {
      const float a  = av[j];
      const float v  = copysignf(expm1f(fabsf(a)), a);   // sign(a)*expm1(|a|)
      const float v2 = v * v;
      psum += v; psum2 += v2;
      int cm = mv[j]; cm = cm < 0 ? 0 : (cm > VM - 1 ? VM - 1 : cm);
      atomicAdd(&h_m_cnt[cm], 1.0f);
      atomicAdd(&h_m_sv[cm],  v);
      atomicAdd(&h_m_sv2[cm], v2);
      int cr = rv[j]; cr = cr < 0 ? 0 : (cr > VT - 1 ? VT - 1 : cr);
      atomicAdd(&h_t_cnt[cr], 1.0f);
      atomicAdd(&h_t_sv[cr],  v);
      atomicAdd(&h_t_sv2[cr], v2);
    }
  }

  // Deterministic wave32 shuffle reduction, then 8 per-wave partials in LDS.
#pragma unroll
  for (int off = 16; off > 0; off >>= 1) {
    psum  += __shfl_down(psum,  off, 32);
    psum2 += __shfl_down(psum2, off, 32);
  }
  if ((t & 31) == 0) { s_red[t >> 5] = psum; s_red2[t >> 5] = psum2; }
  __syncthreads();   // also orders all histogram ds_add_f32 before finalize reads

  float* orow = out + (size_t)b * NCOLS;

  // mcc bins: count [4,260), mean [260,516), std [516,772)
  {
    const float cnt = h_m_cnt[t];
    const float sv  = h_m_sv[t];
    const float sv2 = h_m_sv2[t];
    const float ec  = (t == 0) ? 0.f : cnt;      // category 0 masked out
    const float mean = sv / (ec + EPSF);
    const float avv  = fmaxf(sv2 - sv * sv / (ec + EPSF), 0.f);
    const float stdv = sqrtf(avv / (fmaxf(ec - 1.f, 0.f) + EPSF));
    orow[4 + t]          = ec;
    orow[4 + VM + t]     = mean;
    orow[4 + 2 * VM + t] = stdv;
    if (ec > 0.f) atomicAdd(&s_dist[0], 1.0f);   // exact 0/1 adds: order-independent
  }
  // tr bins: [772,900), [900,1028), [1028,1156)
  if (t < VT) {
    const float cnt = h_t_cnt[t];
    const float sv  = h_t_sv[t];
    const float sv2 = h_t_sv2[t];
    const float ec  = (t == 0) ? 0.f : cnt;
    const float mean = sv / (ec + EPSF);
    const float avv  = fmaxf(sv2 - sv * sv / (ec + EPSF), 0.f);
    const float stdv = sqrtf(avv / (fmaxf(ec - 1.f, 0.f) + EPSF));
    orow[4 + 3 * VM + t]          = ec;
    orow[4 + 3 * VM + VT + t]     = mean;
    orow[4 + 3 * VM + 2 * VT + t] = stdv;
    if (ec > 0.f) atomicAdd(&s_dist[1], 1.0f);
  }
  __syncthreads();

  if (t == 0) {
    float sum = 0.f, sum2 = 0.f;
#pragma unroll
    for (int w = 0; w < 8; ++w) { sum += s_red[w]; sum2 += s_red2[w]; }
    const float sl   = (float)seq_lens[b];
    const float mean = sum / (sl + EPSF);
    const float aa   = fmaxf(sum2 - sum * sum / (sl + EPSF), 0.f);
    const float stdv = sqrtf(aa / (fmaxf(sl - 1.f, 0.f) + EPSF));
    orow[0] = sl; orow[1] = sum; orow[2] = mean; orow[3] = stdv;
    orow[1156] = s_dist[0];
    orow[1157] = s_dist[1];
  }
}

extern "C" void kernel_launch(void* const* d_in, const int* in_sizes, int n_in,
                              void* d_out, int out_size, void* d_ws, size_t ws_size,
                              hipStream_t stream) {
  const float* amount   = (const float*)d_in[0];
  const int*   mcc      = (const int*)d_in[1];
  const int*   trt      = (const int*)d_in[2];
  const int*   seq_lens = (const int*)d_in[3];
  float* out = (float*)d_out;
  const int B = in_sizes[3];          // seq_lens has one entry per row
  agg_feature_kernel<<<B, 256, 0, stream>>>(amount, mcc, trt, seq_lens, out);
}